// pyPD_43602507989353
// MI455X (gfx1250) — compile-verified
//
#include <hip/hip_runtime.h>
#include <hip/hip_bf16.h>

// CDNA5 (gfx1250) projective-dynamics step.
//  - Li [128,1024,1024] never materialized: consumed only via scalar/row sums
//    through the per-element row-sums s[e,k] = sum_n Ji[e,k,n].
//  - A = diag(m) + c*ones*ones^T  =>  exact Sherman-Morrison solve, no Cholesky.
//  - f32 WMMA (V_WMMA_F32_16X16X4_F32) for the per-element [3,1024]x[1024,3]
//    contractions (Dmi = GSi@x in K1, F = Ji@xt in K2), rows padded 3->16.
//  - Inner loop is branchless: lanes with row>=3 feed only unread D positions,
//    so they load clamped-address (finite) garbage unconditionally.
//  - K remapped so each macro-chunk of 16 n-values = 2 global b128 + 2 ds b128
//    loads feeding 4 back-to-back WMMAs (waits amortized, no exec churn).
//  - K2 is one persistent 1024-thread workgroup (n==1024): xt kept transposed
//    in LDS ([3][1024], contiguous B-operand rows); all 5 iterations in-kernel.

typedef __attribute__((ext_vector_type(2))) float v2f;
typedef __attribute__((ext_vector_type(8))) float v8f;

#define N_PTS 1024
#define N_ELEM 128
#define N_ROWS (3 * N_ELEM) // 384 Ji rows

// 4 WMMAs consuming 16 contiguous n-values.
// Lane layout (ISA 16x16x4 f32): lanes 0-15 supply K=0,1; lanes 16-31 K=2,3.
// We map, for sub-chunk j: K0,K1 <- n = base+2j,+1 (lanes<16, half=0),
//                          K2,K3 <- n = base+8+2j,+1 (lanes>=16, half=8),
// so each lane reads 8 contiguous floats per operand per macro-chunk.
__device__ __forceinline__ v8f wmma4(const float4 a0, const float4 a1,
                                     const float4 b0, const float4 b1, v8f acc) {
  v2f a, b;
  a.x = a0.x; a.y = a0.y; b.x = b0.x; b.y = b0.y;
  acc = __builtin_amdgcn_wmma_f32_16x16x4_f32(false, a, false, b, (short)0, acc, false, false);
  a.x = a0.z; a.y = a0.w; b.x = b0.z; b.y = b0.w;
  acc = __builtin_amdgcn_wmma_f32_16x16x4_f32(false, a, false, b, (short)0, acc, false, false);
  a.x = a1.x; a.y = a1.y; b.x = b1.x; b.y = b1.y;
  acc = __builtin_amdgcn_wmma_f32_16x16x4_f32(false, a, false, b, (short)0, acc, false, false);
  a.x = a1.z; a.y = a1.w; b.x = b1.z; b.y = b1.w;
  acc = __builtin_amdgcn_wmma_f32_16x16x4_f32(false, a, false, b, (short)0, acc, false, false);
  return acc;
}

// ---------------------------------------------------------------------------
// Kernel 1: per element e (one block each):
//   Dmi[e] = GSi[e] @ x               (WMMA over n)
//   Ji[e]  = inv(Dmi[e])^T @ GSi[e]   -> workspace [384,1024]
//   sE[e*3+k] = sum_n Ji[e,k,n]       -> workspace [384]
// GSi[e,g,n] = S[4e+g,n] - S[4e+3,n] (G applied implicitly).
// ---------------------------------------------------------------------------
__global__ __launch_bounds__(256) void k_build_ji(const float* __restrict__ x,
                                                  const float* __restrict__ S,
                                                  float* __restrict__ Ji,
                                                  float* __restrict__ sE) {
  const int e    = blockIdx.x;
  const int tid  = threadIdx.x;
  const int lane = tid & 31;
  const int wave = tid >> 5; // 0..7

  __shared__ __align__(16) float sxT[3][N_PTS]; // x transposed (B operand rows)
  __shared__ float ldsPart[8][9];               // per-wave partial Dmi
  __shared__ float ldsBk[9];                    // Bk[k*3+g] = inv(Dmi)[g][k]
  __shared__ float red[256];

  const float* S0 = S + (size_t)(4 * e) * N_PTS;

  // stage x transposed into LDS (coalesced global reads)
  for (int n = tid; n < N_PTS; n += 256) {
    sxT[0][n] = x[n * 3 + 0];
    sxT[1][n] = x[n * 3 + 1];
    sxT[2][n] = x[n * 3 + 2];
  }
  __syncthreads();

  const int row  = lane & 15;               // M for A, N for B
  const int ar   = (row < 3) ? row : 0;     // clamped address row (see header)
  const int half = (lane >> 4) << 3;        // lanes 0-15: 0, lanes 16-31: 8

  const float* Arow = S0 + (size_t)ar * N_PTS;
  const float* A3   = S0 + (size_t)3 * N_PTS;

  // ---- Dmi via WMMA: A[M=g][K=n] = S[g]-S[3], B[K=n][N=d] = x[n][d]
  v8f acc = {0.f, 0.f, 0.f, 0.f, 0.f, 0.f, 0.f, 0.f};
  for (int c = wave; c < N_PTS / 16; c += 8) { // 64 macro-chunks, wave-uniform
    const int base = c * 16 + half;
    const float4 r0 = *(const float4*)(Arow + base);
    const float4 r1 = *(const float4*)(Arow + base + 4);
    const float4 t0 = *(const float4*)(A3 + base);
    const float4 t1 = *(const float4*)(A3 + base + 4);
    float4 a0, a1;
    a0.x = r0.x - t0.x; a0.y = r0.y - t0.y; a0.z = r0.z - t0.z; a0.w = r0.w - t0.w;
    a1.x = r1.x - t1.x; a1.y = r1.y - t1.y; a1.z = r1.z - t1.z; a1.w = r1.w - t1.w;
    const float4 b0 = *(const float4*)(&sxT[ar][base]);
    const float4 b1 = *(const float4*)(&sxT[ar][base + 4]);
    acc = wmma4(a0, a1, b0, b1, acc);
  }
  // D[M][N]: VGPR r holds M=r for lanes 0-15. Dmi[g][d] at lane==d (<3), r==g.
  if (lane < 3) {
    for (int g = 0; g < 3; ++g) ldsPart[wave][g * 3 + lane] = acc[g];
  }
  __syncthreads();

  if (tid == 0) {
    float D[9];
    for (int i = 0; i < 9; ++i) {
      float s = 0.f;
      for (int w = 0; w < 8; ++w) s += ldsPart[w][i];
      D[i] = s;
    }
    const float c00 = D[4] * D[8] - D[5] * D[7];
    const float c01 = -(D[3] * D[8] - D[5] * D[6]);
    const float c02 = D[3] * D[7] - D[4] * D[6];
    const float det = D[0] * c00 + D[1] * c01 + D[2] * c02;
    const float id = 1.0f / det;
    float inv[9];
    inv[0] = c00 * id;
    inv[1] = -(D[1] * D[8] - D[2] * D[7]) * id;
    inv[2] = (D[1] * D[5] - D[2] * D[4]) * id;
    inv[3] = c01 * id;
    inv[4] = (D[0] * D[8] - D[2] * D[6]) * id;
    inv[5] = -(D[0] * D[5] - D[2] * D[3]) * id;
    inv[6] = c02 * id;
    inv[7] = -(D[0] * D[7] - D[1] * D[6]) * id;
    inv[8] = (D[0] * D[4] - D[1] * D[3]) * id;
    for (int k = 0; k < 3; ++k)
      for (int g = 0; g < 3; ++g) ldsBk[k * 3 + g] = inv[g * 3 + k];
  }
  __syncthreads();

  // ---- Ji rows + row sums
  float r0 = 0.f, r1 = 0.f, r2 = 0.f;
  for (int n = tid; n < N_PTS; n += 256) {
    const float s3 = S0[3 * N_PTS + n];
    const float g0 = S0[0 * N_PTS + n] - s3;
    const float g1 = S0[1 * N_PTS + n] - s3;
    const float g2 = S0[2 * N_PTS + n] - s3;
    const float j0 = ldsBk[0] * g0 + ldsBk[1] * g1 + ldsBk[2] * g2;
    const float j1 = ldsBk[3] * g0 + ldsBk[4] * g1 + ldsBk[5] * g2;
    const float j2 = ldsBk[6] * g0 + ldsBk[7] * g1 + ldsBk[8] * g2;
    Ji[(size_t)(e * 3 + 0) * N_PTS + n] = j0;
    Ji[(size_t)(e * 3 + 1) * N_PTS + n] = j1;
    Ji[(size_t)(e * 3 + 2) * N_PTS + n] = j2;
    r0 += j0;
    r1 += j1;
    r2 += j2;
  }
  for (int comp = 0; comp < 3; ++comp) {
    red[tid] = (comp == 0) ? r0 : ((comp == 1) ? r1 : r2);
    __syncthreads();
    for (int s = 128; s > 0; s >>= 1) {
      if (tid < s) red[tid] += red[tid + s];
      __syncthreads();
    }
    if (tid == 0) sE[e * 3 + comp] = red[0];
    __syncthreads();
  }
}

// ---------------------------------------------------------------------------
// 3x3 Jacobi eigensolver helper (on symmetric A, accumulates V).
// ---------------------------------------------------------------------------
__device__ inline void jacobi_rot(float A[3][3], float V[3][3], int p, int q) {
  const float apq = A[p][q];
  if (fabsf(apq) < 1e-30f) return;
  const float tau = (A[q][q] - A[p][p]) / (2.0f * apq);
  const float t = (tau >= 0.f ? 1.f : -1.f) / (fabsf(tau) + sqrtf(1.f + tau * tau));
  const float c = 1.0f / sqrtf(1.f + t * t);
  const float s = t * c;
  for (int k = 0; k < 3; ++k) {
    const float akp = A[k][p], akq = A[k][q];
    A[k][p] = c * akp - s * akq;
    A[k][q] = s * akp + c * akq;
  }
  for (int k = 0; k < 3; ++k) {
    const float apk = A[p][k], aqk = A[q][k];
    A[p][k] = c * apk - s * aqk;
    A[q][k] = s * apk + c * aqk;
  }
  for (int k = 0; k < 3; ++k) {
    const float vkp = V[k][p], vkq = V[k][q];
    V[k][p] = c * vkp - s * vkq;
    V[k][q] = s * vkp + c * vkq;
  }
}

// ---------------------------------------------------------------------------
// Kernel 2: persistent solver. One workgroup, 1024 threads (one per point).
// Init: masses=diag(M) (regs), y = x + dt*v (regs), xt = y (LDS, transposed),
//       Li_colsum (reg per thread), sumLi, trD (LDS tree).
// 5 iterations: F=Ji@xt (WMMA) -> 3x3 SVD -> scalars s1,s2 ->
//               gf -> Sherman-Morrison update of xt.  Output xt.
// ---------------------------------------------------------------------------
__global__ __launch_bounds__(1024, 1) void k_solve(
    const float* __restrict__ x, const float* __restrict__ v,
    const float* __restrict__ Mmat, const float* __restrict__ dtp,
    const float* __restrict__ Ji, const float* __restrict__ sE,
    float* __restrict__ out) {
  const int tid  = threadIdx.x; // 0..1023
  const int lane = tid & 31;
  const int wave = tid >> 5;    // 0..31

  __shared__ __align__(16) float sXtT[3][N_PTS]; // xt transposed (B rows)
  __shared__ float sF[N_ELEM * 9];
  __shared__ float red[N_PTS];
  __shared__ float sScal[4];

  const float dt = dtp[0];
  const float dt2 = dt * dt;

  // ---- per-point state in registers (each row owned by exactly one thread)
  const float m = Mmat[(size_t)tid * (N_PTS + 1)]; // diagonal of M
  const float im = 1.0f / m;
  float yv[3];
  for (int d = 0; d < 3; ++d) {
    yv[d] = x[tid * 3 + d] + dt * v[tid * 3 + d];
    sXtT[d][tid] = yv[d]; // xt = y
  }
  // Li_colsum[tid] = sum_r sE[r] * Ji[r, tid]
  float cs = 0.f;
  for (int r = 0; r < N_ROWS; ++r) cs += sE[r] * Ji[(size_t)r * N_PTS + tid];

  // trD = sum 1/m
  red[tid] = im;
  __syncthreads();
  for (int s = 512; s > 0; s >>= 1) {
    if (tid < s) red[tid] += red[tid + s];
    __syncthreads();
  }
  if (tid == 0) sScal[0] = red[0];
  __syncthreads();
  // sumLi = sum_r sE[r]^2
  red[tid] = (tid < N_ROWS) ? sE[tid] * sE[tid] : 0.f;
  __syncthreads();
  for (int s = 512; s > 0; s >>= 1) {
    if (tid < s) red[tid] += red[tid + s];
    __syncthreads();
  }
  if (tid == 0) sScal[1] = red[0];
  __syncthreads();
  const float cc = dt2 * sScal[1];          // c in A = D + c*11^T
  const float denom = 1.0f + cc * sScal[0]; // 1 + c * tr(D^-1)
  __syncthreads();

  const int row  = lane & 15;
  const int ar   = (row < 3) ? row : 0; // clamped address row, branchless
  const int half = (lane >> 4) << 3;

  for (int it = 0; it < 5; ++it) {
    // ---- F[e] = Ji[e] @ xt via WMMA (each wave: 4 elements, uniform flow)
    for (int e = wave; e < N_ELEM; e += 32) {
      const float* Arow = Ji + (size_t)(e * 3 + ar) * N_PTS;
      __builtin_prefetch(Arow, 0, 1);
      v8f acc = {0.f, 0.f, 0.f, 0.f, 0.f, 0.f, 0.f, 0.f};
      for (int c = 0; c < N_PTS / 16; ++c) {
        const int base = c * 16 + half;
        const float4 a0 = *(const float4*)(Arow + base);
        const float4 a1 = *(const float4*)(Arow + base + 4);
        const float4 b0 = *(const float4*)(&sXtT[ar][base]);
        const float4 b1 = *(const float4*)(&sXtT[ar][base + 4]);
        acc = wmma4(a0, a1, b0, b1, acc);
      }
      if (lane < 3) {
        for (int g = 0; g < 3; ++g) sF[e * 9 + g * 3 + lane] = acc[g];
      }
    }
    __syncthreads();

    // ---- per-element 3x3 SVD, R = U^T V, s2 contribution
    float s2c = 0.f;
    if (tid < N_ELEM) {
      const int e = tid;
      float Fm[3][3];
      for (int g = 0; g < 3; ++g)
        for (int d = 0; d < 3; ++d) Fm[g][d] = sF[e * 9 + g * 3 + d];
      float A[3][3], V[3][3];
      for (int i = 0; i < 3; ++i)
        for (int j = 0; j < 3; ++j) {
          float s = 0.f;
          for (int g = 0; g < 3; ++g) s += Fm[g][i] * Fm[g][j];
          A[i][j] = s;
          V[i][j] = (i == j) ? 1.f : 0.f;
        }
      for (int sweep = 0; sweep < 4; ++sweep) {
        jacobi_rot(A, V, 0, 1);
        jacobi_rot(A, V, 0, 2);
        jacobi_rot(A, V, 1, 2);
      }
      float lam[3] = {A[0][0], A[1][1], A[2][2]};
      for (int i = 0; i < 2; ++i) // sort descending (LAPACK convention)
        for (int j = i + 1; j < 3; ++j)
          if (lam[j] > lam[i]) {
            const float tl = lam[i]; lam[i] = lam[j]; lam[j] = tl;
            for (int r2 = 0; r2 < 3; ++r2) {
              const float tv = V[r2][i]; V[r2][i] = V[r2][j]; V[r2][j] = tv;
            }
          }
      float U[3][3];
      for (int i = 0; i < 3; ++i) {
        const float sig = sqrtf(fmaxf(lam[i], 0.f));
        const float is = (sig > 1e-12f) ? 1.f / sig : 0.f;
        for (int r2 = 0; r2 < 3; ++r2)
          U[r2][i] = (Fm[r2][0] * V[0][i] + Fm[r2][1] * V[1][i] +
                      Fm[r2][2] * V[2][i]) * is;
      }
      // s2 += sum_k sE[e,k] * sum_d R[k][d],  R[k][d] = U_col_k . V_col_d
      for (int k = 0; k < 3; ++k) {
        float rs = 0.f;
        for (int d = 0; d < 3; ++d) {
          float rkd = 0.f;
          for (int r2 = 0; r2 < 3; ++r2) rkd += U[r2][k] * V[r2][d];
          rs += rkd;
        }
        s2c += sE[e * 3 + k] * rs;
      }
    }

    // ---- (s1 - s2) in one fused tree reduction
    const float xs = sXtT[0][tid] + sXtT[1][tid] + sXtT[2][tid];
    red[tid] = cs * xs - ((tid < N_ELEM) ? s2c : 0.f);
    __syncthreads();
    for (int s = 512; s > 0; s >>= 1) {
      if (tid < s) red[tid] += red[tid + s];
      __syncthreads();
    }
    if (tid == 0) sScal[3] = red[0];
    __syncthreads();
    const float s12 = sScal[3];

    // ---- gf, q = D^-1 gf, then the three column sums p_d
    float q[3];
    for (int d = 0; d < 3; ++d) {
      const float gf = m * (sXtT[d][tid] - yv[d]) + dt2 * s12;
      q[d] = gf * im;
    }
    for (int d = 0; d < 3; ++d) {
      red[tid] = q[d];
      __syncthreads();
      for (int s = 512; s > 0; s >>= 1) {
        if (tid < s) red[tid] += red[tid + s];
        __syncthreads();
      }
      if (tid == 0) sScal[d] = red[0];
      __syncthreads();
    }

    // ---- Sherman-Morrison: xt_i -= q_i - (c*p_d/denom)/m_i
    const float corr = cc / denom;
    for (int d = 0; d < 3; ++d)
      sXtT[d][tid] -= (q[d] - corr * sScal[d] * im);
    __syncthreads();
  }

  for (int d = 0; d < 3; ++d) out[tid * 3 + d] = sXtT[d][tid];
}

// ---------------------------------------------------------------------------
extern "C" void kernel_launch(void* const* d_in, const int* in_sizes, int n_in,
                              void* d_out, int out_size, void* d_ws,
                              size_t ws_size, hipStream_t stream) {
  const float* x = (const float*)d_in[0];  // [1024,3]
  const float* v = (const float*)d_in[1];  // [1024,3]
  const float* S = (const float*)d_in[2];  // [512,1024]
  const float* M = (const float*)d_in[3];  // [1024,1024]
  const float* dt = (const float*)d_in[4]; // scalar
  float* out = (float*)d_out;              // [1024,3]

  float* ws = (float*)d_ws;
  float* Ji = ws;                          // [384,1024] = 1.5 MB
  float* sE = ws + (size_t)N_ROWS * N_PTS; // [384]

  k_build_ji<<<N_ELEM, 256, 0, stream>>>(x, S, Ji, sE);
  k_solve<<<1, N_PTS, 0, stream>>>(x, v, M, dt, Ji, sE, out);
}